// TGCNCell_66099546685627
// MI455X (gfx1250) — compile-verified
//
#include <hip/hip_runtime.h>
#include <math.h>

typedef __attribute__((ext_vector_type(2))) float v2f;
typedef __attribute__((ext_vector_type(8))) float v8f;

#define IN_FEATS 32
#define GATES 128   // 4 * OUT_FEATS
#define OUT_FEATS 32

// ---------------------------------------------------------------------------
// Fast activations: use the CDNA5 native V_TANH_F32 transcendental when the
// toolchain exposes it; otherwise a v_exp_f32 + v_rcp_f32 formulation.
// sigmoid(x) = 0.5*tanh(x/2) + 0.5  -> 1 trans op + 2 FMA-class ops.
// ---------------------------------------------------------------------------
__device__ __forceinline__ float fast_tanh(float x) {
#if __has_builtin(__builtin_amdgcn_tanhf)
    return __builtin_amdgcn_tanhf(x);
#else
    // tanh(x) = 1 - 2/(exp(2x)+1)
    float e = __expf(2.0f * x);
    return 1.0f - 2.0f * __builtin_amdgcn_rcpf(e + 1.0f);
#endif
}

__device__ __forceinline__ float fast_sigmoid(float x) {
    return __builtin_fmaf(0.5f, fast_tanh(0.5f * x), 0.5f);
}

// ---------------------------------------------------------------------------
// Kernel 1: zero the aggregation buffer (must run every launch; harness does
// not re-poison workspace between replays).
// ---------------------------------------------------------------------------
__global__ void tgcn_zero_kernel(float* __restrict__ agg, int n) {
    int i = blockIdx.x * blockDim.x + threadIdx.x;
    if (i < n) agg[i] = 0.0f;
}

// ---------------------------------------------------------------------------
// Kernel 2: edge scatter-add.  One wave32 per edge, lane == feature column.
// x[src] gather is 128B coalesced per wave; the 12.8MB destination lives in
// L2 (192MB), so the f32 atomics resolve there.
// ---------------------------------------------------------------------------
__global__ void tgcn_scatter_kernel(const float* __restrict__ x,
                                    const int*   __restrict__ src,
                                    const int*   __restrict__ dst,
                                    float*       __restrict__ agg,
                                    int nEdges) {
    unsigned tid  = blockIdx.x * blockDim.x + threadIdx.x;
    unsigned e    = tid >> 5;          // edge id (one wave per edge)
    unsigned lane = tid & 31u;         // feature column
    if (e >= (unsigned)nEdges) return;
    int s = src[e];
    int d = dst[e];
    float v = x[(size_t)s * IN_FEATS + lane];
    atomicAdd(&agg[(size_t)d * IN_FEATS + lane], v);
}

// ---------------------------------------------------------------------------
// Kernel 3: fused gates GEMM (h_agg @ W_ih^T + b_ih + b_hh) + LSTM-style
// activations, using V_WMMA_F32_16X16X4_F32.
//
// Per wave: M-tile of 16 nodes, K=32 (8 wmma k-steps of 4), N covered by
// six 16-wide tiles {0,1, 4,5, 6,7} (i, g, o gates; the f gate is dead in
// the reference).  fp32 WMMA keeps full reference precision.
//
// A 16x4 f32 layout (ISA 7.12.2): lane m=lane&15; VGPR pair holds
// K = k0 + 2*(lane>>4) and K+1.  B 4x16 mirrors it: lane n=lane&15,
// rows K = k0 + 2*(lane>>4), K+1.  C/D: VGPR r, lanes0-15 -> M=r,N=lane;
// lanes16-31 -> M=r+8, N=lane-16.
// ---------------------------------------------------------------------------
__global__ void __launch_bounds__(256)
tgcn_gates_kernel(const float* __restrict__ agg,
                  const float* __restrict__ W_ih,   // [128, 32] row-major
                  const float* __restrict__ b_ih,   // [128]
                  const float* __restrict__ b_hh,   // [128]
                  float*       __restrict__ out,    // [nNodes, 32]
                  int nNodes) {
    __shared__ float sW[GATES * IN_FEATS];  // 16KB: W_ih staged row-major
    __shared__ float sB[GATES];             // b_ih + b_hh

    // Stage weights + bias into LDS.
    for (int i = threadIdx.x; i < GATES * IN_FEATS; i += 256)
        sW[i] = W_ih[i];
    if (threadIdx.x < GATES)
        sB[threadIdx.x] = b_ih[threadIdx.x] + b_hh[threadIdx.x];
    __syncthreads();

    const int wave  = threadIdx.x >> 5;          // 0..7
    const int lane  = threadIdx.x & 31;
    const int half  = lane >> 4;                 // 0 or 1
    const int mn    = lane & 15;                 // row (A) / col (B,C)
    const int kr    = half * 2;                  // K sub-offset within a k-step
    const int mBase = (blockIdx.x * 8 + wave) * 16;

    // ---- preload A: this lane's half-row of h_agg (8 float2 = 16 floats) ----
    int rowA = mBase + mn;
    if (rowA > nNodes - 1) rowA = nNodes - 1;    // clamp; tail stores are guarded
    const float* Ar = agg + (size_t)rowA * IN_FEATS + kr;
    v2f a[8];
#pragma unroll
    for (int k = 0; k < 8; ++k)
        a[k] = *reinterpret_cast<const v2f*>(Ar + 4 * k);   // K = 4k+kr, 4k+kr+1

    // ---- accumulate the six live N-tiles: i(0,1) g(4,5) o(6,7) ----
    const int tn[6] = {0, 1, 4, 5, 6, 7};
    v8f acc[6];
#pragma unroll
    for (int t = 0; t < 6; ++t) {
        const int nCol = tn[t] * 16 + mn;
        const float bv = sB[nCol];
        v8f c = {bv, bv, bv, bv, bv, bv, bv, bv};   // bias broadcast down M
        const float* Wr = &sW[nCol * IN_FEATS + kr];
#pragma unroll
        for (int k = 0; k < 8; ++k) {
            v2f b = *reinterpret_cast<const v2f*>(Wr + 4 * k);  // B[K][n]=W_ih[n][K]
            c = __builtin_amdgcn_wmma_f32_16x16x4_f32(
                    /*neg_a=*/false, a[k],
                    /*neg_b=*/false, b,
                    /*c_mod=*/(short)0, c,
                    /*reuse_a=*/false, /*reuse_b=*/false);
        }
        acc[t] = c;
    }

    // ---- activations + store ----
    // acc idx: i -> {0,1}, g -> {2,3}, o -> {4,5}; t2 selects col group 0..15/16..31
#pragma unroll
    for (int t2 = 0; t2 < 2; ++t2) {
#pragma unroll
        for (int r = 0; r < 8; ++r) {
            int row = mBase + r + 8 * half;
            float iv = acc[0 + t2][r];
            float gv = acc[2 + t2][r];
            float ov = acc[4 + t2][r];
            float cc = fast_sigmoid(iv) * fast_tanh(gv);
            float hv = fast_sigmoid(ov) * fast_tanh(cc);
            hv = fmaxf(hv, 0.0f);                // relu
            if (row < nNodes)
                out[(size_t)row * OUT_FEATS + t2 * 16 + mn] = hv;
        }
    }
}

// ---------------------------------------------------------------------------
extern "C" void kernel_launch(void* const* d_in, const int* in_sizes, int n_in,
                              void* d_out, int out_size, void* d_ws, size_t ws_size,
                              hipStream_t stream) {
    const float* x    = (const float*)d_in[0];
    const int*   src  = (const int*)  d_in[1];
    const int*   dst  = (const int*)  d_in[2];
    const float* W_ih = (const float*)d_in[3];
    // d_in[4] = W_hh: unused by the reference math (only its shape is read)
    const float* b_ih = (const float*)d_in[5];
    const float* b_hh = (const float*)d_in[6];
    float* out = (float*)d_out;

    const int nNodes = in_sizes[0] / IN_FEATS;
    const int nEdges = in_sizes[1];

    const size_t aggBytes = (size_t)nNodes * IN_FEATS * sizeof(float);
    // Prefer workspace; fall back to in-place accumulation in d_out (safe:
    // each wave reads only the rows it later overwrites).
    float* agg = (ws_size >= aggBytes) ? (float*)d_ws : out;

    // 1) zero accumulator
    {
        int n = nNodes * IN_FEATS;
        tgcn_zero_kernel<<<(n + 255) / 256, 256, 0, stream>>>(agg, n);
    }
    // 2) edge scatter-add (one wave per edge)
    {
        unsigned total = (unsigned)nEdges * 32u;
        tgcn_scatter_kernel<<<(total + 255u) / 256u, 256, 0, stream>>>(
            x, src, dst, agg, nEdges);
    }
    // 3) fused WMMA GEMM + gate activations (128 nodes per 256-thread block)
    {
        int blocks = (nNodes + 127) / 128;
        tgcn_gates_kernel<<<blocks, 256, 0, stream>>>(
            agg, W_ih, b_ih, b_hh, out, nNodes);
    }
}